// LIFNeuron_10557029614159
// MI455X (gfx1250) — compile-verified
//
#include <hip/hip_runtime.h>

// LIF neuron scan, MI455X (gfx1250).
// Shapes fixed by the reference: x is [T*B, C, H, W] = [8*32, 128, 32, 32] fp32.
// Memory-bound streaming kernel: b128 NT loads/stores, 8-deep load MLP per thread.

#define T_STEPS   8
#define N_PER_T   (32 * 128 * 32 * 32)   // B*C*H*W = 4,194,304 elements per timestep
#define N_VEC4    (N_PER_T / 4)          // 1,048,576 float4 work items
#define BLOCK_SZ  256

typedef __attribute__((ext_vector_type(4))) float v4f;

__global__ __launch_bounds__(BLOCK_SZ) void lif_scan_kernel(
    const float* __restrict__ x,
    const float* __restrict__ vth_ptr,
    float* __restrict__ out)
{
    const int i = blockIdx.x * BLOCK_SZ + threadIdx.x;
    if (i >= N_VEC4) return;

    // ---- uniform scalar preamble (SALU / s_load_b32) ----
    // no-grad clamp: relu(Vth - bound) + bound
    float vth = vth_ptr[0];
    vth = fmaxf(vth - 0.0005f, 0.0f) + 0.0005f;
    const float thr     = 0.3f * vth;            // ALPHA * Vth
    const float inv_dt  = 20.0f;                 // 1 / DELTA_T
    const float beta    = 0.9512294530868530273f;  // expf(-DELTA_T/TAU)
    const float one_mb  = 1.0f - beta;

    const v4f* __restrict__ xv = (const v4f*)x;
    v4f* __restrict__ ov       = (v4f*)out;

    // ---- issue all 8 timestep loads up front: 8 outstanding b128 NT loads ----
    v4f xt[T_STEPS];
#pragma unroll
    for (int t = 0; t < T_STEPS; ++t) {
        xt[t] = __builtin_nontemporal_load(&xv[t * N_VEC4 + i]);
    }

    // ---- register-resident scan over T ----
    v4f mem;
    mem.x = 0.0f; mem.y = 0.0f; mem.z = 0.0f; mem.w = 0.0f;

#pragma unroll
    for (int t = 0; t < T_STEPS; ++t) {
        // leaky integrate: mem = beta*mem + (1-beta)*x[t]
#pragma unroll
        for (int c = 0; c < 4; ++c) {
            mem[c] = __builtin_fmaf(beta, mem[c], one_mb * xt[t][c]);
        }
        // threshold, reset-by-subtraction, scale by 1/dt
        v4f spk;
#pragma unroll
        for (int c = 0; c < 4; ++c) {
            const float s = (mem[c] >= thr) ? vth : 0.0f;
            mem[c] -= s;
            spk[c] = s * inv_dt;
        }
        __builtin_nontemporal_store(spk, &ov[t * N_VEC4 + i]);
    }
}

extern "C" void kernel_launch(void* const* d_in, const int* in_sizes, int n_in,
                              void* d_out, int out_size, void* d_ws, size_t ws_size,
                              hipStream_t stream)
{
    const float* x   = (const float*)d_in[0];
    const float* vth = (const float*)d_in[1];
    float* out       = (float*)d_out;

    dim3 block(BLOCK_SZ);
    dim3 grid(N_VEC4 / BLOCK_SZ);   // 4096 blocks, exact cover
    lif_scan_kernel<<<grid, block, 0, stream>>>(x, vth, out);
}